// EntityClassify_HeteroAPI_91259465105420
// MI455X (gfx1250) — compile-verified
//
#include <hip/hip_runtime.h>

#define NA      100000
#define NB_NODE 100000
#define NREL    6
#define EPR     500000
#define HD      64
#define OD      16
#define NBASES  4

typedef __attribute__((ext_vector_type(2))) float v2f;
typedef __attribute__((ext_vector_type(8))) float v8f;

// REL_SRC = A,B,A,B,A,B  -> src is A iff r even
// REL_DST = B,A,A,B,B,A

// ---------------- degree counting / reciprocal ----------------
__global__ __launch_bounds__(256) void k_count_deg(const int* __restrict__ dst,
                                                   float* __restrict__ deg, int n) {
  int i = blockIdx.x * blockDim.x + threadIdx.x;
  if (i < n) unsafeAtomicAdd(&deg[dst[i]], 1.0f);
}

__global__ __launch_bounds__(256) void k_recip(float* __restrict__ deg, int n) {
  int i = blockIdx.x * blockDim.x + threadIdx.x;
  if (i < n) deg[i] = 1.0f / fmaxf(deg[i], 1.0f);
}

// ---------------- per-edge gather + normalized scatter-add ----------------
// All offsets fit in 32 bits (max 6.4M floats), so use int indexing to keep
// address math on 32-bit VALU + saddr-form global ops.
template <int D>
__global__ __launch_bounds__(256) void k_scatter(const float* __restrict__ X,
                                                 const int* __restrict__ src,
                                                 const int* __restrict__ dst,
                                                 const float* __restrict__ inv,
                                                 float* __restrict__ out, int nE) {
  const int QP = D / 4;  // float4 quads per edge row
  int gid = blockIdx.x * blockDim.x + threadIdx.x;   // <= 8M, fits int
  if (gid >= nE * QP) return;
  int e = gid / QP;
  int q = gid % QP;
  int s = src[e];
  int d = dst[e];
  float w = inv[d];
  const float4 v = *(const float4*)(X + s * D + q * 4);
  float* o = out + d * D + q * 4;
  unsafeAtomicAdd(o + 0, v.x * w);
  unsafeAtomicAdd(o + 1, v.y * w);
  unsafeAtomicAdd(o + 2, v.z * w);
  unsafeAtomicAdd(o + 3, v.w * w);
}

// ---------------- bias (+ optional ReLU), D is a power of two ----------------
__global__ __launch_bounds__(256) void k_bias_act(float* __restrict__ h,
                                                  const float* __restrict__ bias,
                                                  int n, int D, int relu) {
  int i = blockIdx.x * blockDim.x + threadIdx.x;
  if (i < n) {
    float v = h[i] + bias[i & (D - 1)];
    h[i] = relu ? fmaxf(v, 0.0f) : v;
  }
}

// ---------------- W_r = sum_b w_comp[r,b] * basis[b] ----------------
__global__ __launch_bounds__(256) void k_combine(const float* __restrict__ wc,
                                                 const float* __restrict__ basis,
                                                 float* __restrict__ W, int perRel) {
  int i = blockIdx.x * blockDim.x + threadIdx.x;
  if (i >= NREL * perRel) return;
  int r = i / perRel;
  int j = i % perRel;
  float acc = 0.0f;
#pragma unroll
  for (int b = 0; b < NBASES; ++b)
    acc += wc[r * NBASES + b] * basis[b * perRel + j];
  W[i] = acc;
}

// ---------------- WMMA f32 GEMM: C[MxN] = A[Mx64] * W[64xN] ----------------
// M multiple of 16, N in {16,64}, K=64. One wave per 16x16 C tile via
// V_WMMA_F32_16X16X4_F32 (full fp32 precision on the matrix pipe).
// W (<=16KB) is staged once per workgroup into LDS and shared by all 4 waves;
// B fragments come from ds_load instead of redundant strided global loads.
__global__ __launch_bounds__(128) void k_gemm_wmma(const float* __restrict__ A,
                                                   const float* __restrict__ W,
                                                   float* __restrict__ C,
                                                   int M, int N) {
  __shared__ float sW[HD * HD];  // 16 KB max (N=64); N=16 uses first 4 KB

  // Cooperative stage of W[64*N] with float4s: 128 threads x 16B.
  const int nW = HD * N;  // 4096 or 1024
  for (int i = threadIdx.x * 4; i < nW; i += 128 * 4)
    *(float4*)(sW + i) = *(const float4*)(W + i);
  __syncthreads();

  const int lane = threadIdx.x & 31;
  const int wave = threadIdx.x >> 5;
  const int tilesN = N >> 4;
  const int nTiles = (M >> 4) * tilesN;
  const int tile = blockIdx.x * 4 + wave;  // wave-uniform
  if (tile >= nTiles) return;              // whole-wave exit: EXEC stays all-1s
  const int m0 = (tile / tilesN) << 4;
  const int n0 = (tile % tilesN) << 4;
  const int half = lane >> 4;              // 0: K={0,1}, 1: K={2,3}
  const int l16  = lane & 15;

  const float* Arow = A + (m0 + l16) * HD;   // 32-bit offsets throughout
  const float* Wcol = sW + n0 + l16;

  v8f c = {};
#pragma unroll
  for (int k = 0; k < 64; k += 4) {
    const int kk = k + half * 2;
    v2f a = *(const v2f*)(Arow + kk);        // A[row][kk..kk+1], global_load_b64
    v2f b;
    b.x = Wcol[kk * N];                       // W[kk][col] from LDS
    b.y = Wcol[(kk + 1) * N];
    c = __builtin_amdgcn_wmma_f32_16x16x4_f32(
        /*neg_a=*/false, a, /*neg_b=*/false, b,
        /*c_mod=*/(short)0, c, /*reuse_a=*/false, /*reuse_b=*/false);
  }

  // C layout: VGPR r -> row m0 + half*8 + r, col n0 + l16
  float* Crow = C + (m0 + half * 8) * N + n0 + l16;
#pragma unroll
  for (int r = 0; r < 8; ++r)
    Crow[r * N] = c[r];
}

// ---------------------------------------------------------------------------
extern "C" void kernel_launch(void* const* d_in, const int* in_sizes, int n_in,
                              void* d_out, int out_size, void* d_ws, size_t ws_size,
                              hipStream_t stream) {
  (void)in_sizes; (void)n_in; (void)out_size; (void)ws_size;

  const float* embedA = (const float*)d_in[0];
  const float* embedB = (const float*)d_in[1];
  const float* bias0  = (const float*)d_in[2];
  const float* wc1    = (const float*)d_in[3];
  const float* basis1 = (const float*)d_in[4];
  const float* bias1  = (const float*)d_in[5];
  const float* wc2    = (const float*)d_in[6];
  const float* basis2 = (const float*)d_in[7];
  const float* bias2  = (const float*)d_in[8];
  const int* src[NREL];
  const int* dst[NREL];
  for (int r = 0; r < NREL; ++r) {
    src[r] = (const int*)d_in[9 + 2 * r];
    dst[r] = (const int*)d_in[10 + 2 * r];
  }
  const int srcIsA[NREL] = {1, 0, 1, 0, 1, 0};  // A,B,A,B,A,B
  const int dstIsA[NREL] = {0, 1, 1, 0, 0, 1};  // B,A,A,B,B,A

  // ---- workspace bump allocation (floats) ----
  float* ws  = (float*)d_ws;
  float* inv = ws;  ws += (size_t)NREL * NA;          // per-relation 1/deg
  float* h0A = ws;  ws += (size_t)NA * HD;
  float* h0B = ws;  ws += (size_t)NB_NODE * HD;
  float* h1A = ws;  ws += (size_t)NA * HD;
  float* h1B = ws;  ws += (size_t)NB_NODE * HD;
  float* xbuf = ws; ws += (size_t)NA * HD;            // reusable GEMM output (serial per relation)
  float* ybuf = ws; ws += (size_t)NA * OD;
  float* W1  = ws;  ws += (size_t)NREL * HD * HD;
  float* W2  = ws;  ws += (size_t)NREL * HD * OD;

  // ---- zero accumulators (graph-capture-safe async memsets) ----
  hipMemsetAsync(inv, 0, sizeof(float) * (size_t)NREL * NA, stream);
  hipMemsetAsync(h0A, 0, sizeof(float) * (size_t)NA * HD, stream);
  hipMemsetAsync(h0B, 0, sizeof(float) * (size_t)NB_NODE * HD, stream);
  hipMemsetAsync(h1A, 0, sizeof(float) * (size_t)NA * HD, stream);
  hipMemsetAsync(h1B, 0, sizeof(float) * (size_t)NB_NODE * HD, stream);
  hipMemsetAsync(d_out, 0, sizeof(float) * (size_t)NA * OD, stream);

  const int gE     = (EPR + 255) / 256;
  const int gInv   = (NREL * NA + 255) / 256;
  const int gs64   = (EPR * (HD / 4) + 255) / 256;
  const int gs16   = (EPR * (OD / 4) + 255) / 256;
  const int gBias64 = (NA * HD + 255) / 256;
  const int gBias16 = (NA * OD + 255) / 256;
  const int gGemm64 = ((NA / 16) * (HD / 16) + 3) / 4;  // waves/4
  const int gGemm16 = ((NA / 16) * (OD / 16) + 3) / 4;
  const int gW1 = (NREL * HD * HD + 255) / 256;
  const int gW2 = (NREL * HD * OD + 255) / 256;

  // ---- degrees (shared by all 3 layers) ----
  for (int r = 0; r < NREL; ++r)
    k_count_deg<<<gE, 256, 0, stream>>>(dst[r], inv + (size_t)r * NA, EPR);
  k_recip<<<gInv, 256, 0, stream>>>(inv, NREL * NA);

  // ---- layer 0: normalized aggregation of raw embeddings ----
  for (int r = 0; r < NREL; ++r) {
    const float* X = srcIsA[r] ? embedA : embedB;
    float* O       = dstIsA[r] ? h0A : h0B;
    k_scatter<HD><<<gs64, 256, 0, stream>>>(X, src[r], dst[r], inv + (size_t)r * NA, O, EPR);
  }
  k_bias_act<<<gBias64, 256, 0, stream>>>(h0A, bias0, NA * HD, HD, 1);
  k_bias_act<<<gBias64, 256, 0, stream>>>(h0B, bias0, NB_NODE * HD, HD, 1);

  // ---- layer 1: basis-combined 64x64 weights, WMMA GEMM, aggregate ----
  k_combine<<<gW1, 256, 0, stream>>>(wc1, basis1, W1, HD * HD);
  for (int r = 0; r < NREL; ++r) {
    const float* H = srcIsA[r] ? h0A : h0B;
    k_gemm_wmma<<<gGemm64, 128, 0, stream>>>(H, W1 + (size_t)r * HD * HD, xbuf, NA, HD);
    float* O = dstIsA[r] ? h1A : h1B;
    k_scatter<HD><<<gs64, 256, 0, stream>>>(xbuf, src[r], dst[r], inv + (size_t)r * NA, O, EPR);
  }
  k_bias_act<<<gBias64, 256, 0, stream>>>(h1A, bias1, NA * HD, HD, 1);
  k_bias_act<<<gBias64, 256, 0, stream>>>(h1B, bias1, NB_NODE * HD, HD, 1);

  // ---- layer 2: only relations with dst == 'A' contribute to the output ----
  k_combine<<<gW2, 256, 0, stream>>>(wc2, basis2, W2, HD * OD);
  const int relsToA[3] = {1, 2, 5};
  for (int t = 0; t < 3; ++t) {
    int r = relsToA[t];
    const float* H = srcIsA[r] ? h1A : h1B;
    k_gemm_wmma<<<gGemm16, 128, 0, stream>>>(H, W2 + (size_t)r * HD * OD, ybuf, NA, OD);
    k_scatter<OD><<<gs16, 256, 0, stream>>>(ybuf, src[r], dst[r], inv + (size_t)r * NA,
                                            (float*)d_out, EPR);
  }
  k_bias_act<<<gBias16, 256, 0, stream>>>((float*)d_out, bias2, NA * OD, OD, 0);
}